// VladPooling_66194035966256
// MI455X (gfx1250) — compile-verified
//
#include <hip/hip_runtime.h>
#include <hip/hip_bf16.h>

// ---- problem constants -----------------------------------------------------
#define BATCH   32
#define SPOS    800      // 16*50 spatial positions
#define DIM     512
#define KG      10       // total centers
#define KPAD    16       // padded M for WMMA
#define KC      8        // centers kept in output
#define SPLIT   10       // K-split: blocks per batch
#define ROWS    (SPOS / SPLIT)   // 80 rows per block
#define CHUNK   8                // rows staged per LDS pass (double buffered)
#define NCHUNK  (ROWS / CHUNK)   // 10
#define FS      520      // LDS row stride (floats): 2*520 % 64 == 16 -> EXEC halves hit disjoint banks
#define AS      24       // LDS row stride for A tile: 2*24 % 64 == 48 -> disjoint banks

typedef float v2f __attribute__((ext_vector_type(2)));
typedef float v8f __attribute__((ext_vector_type(8)));

// =============================================================================
// Kernel 1: per (batch, K-segment) block: softmax + WMMA partial wf / a_sum.
// feat is streamed HBM -> LDS with async copies (ASYNCcnt), double buffered.
// =============================================================================
__global__ __launch_bounds__(256) void vlad_partial_kernel(
    const float* __restrict__ feat,      // [B][S][D]
    const float* __restrict__ score,     // [B][S][KG]
    float* __restrict__ wsWf,            // [B][KPAD][D]  (atomically accumulated)
    float* __restrict__ wsAsum)          // [B][KPAD]
{
    const int b    = blockIdx.x;
    const int seg  = blockIdx.y;
    const int tid  = threadIdx.x;
    const int wave = tid >> 5;           // 8 waves
    const int lane = tid & 31;
    const int half = lane >> 4;          // EXEC half for WMMA fragment layout
    const int l15  = lane & 15;

    __shared__ float fSm[2 * CHUNK * FS];   // double-buffered 8x512 feat tiles
    __shared__ float aSm[2 * CHUNK * AS];   // double-buffered 8x16 softmax tiles

    v8f acc0 = {}, acc1 = {}, acc2 = {}, acc3 = {};
    float asumAcc = 0.0f;

    const float* fb = feat  + ((size_t)b * SPOS + (size_t)seg * ROWS) * DIM;
    const float* sb = score + ((size_t)b * SPOS + (size_t)seg * ROWS) * KG;

    // ---- async HBM -> LDS stage of one 8x512 chunk (4 x b128 per lane) ----
    auto issueChunk = [&](int ch, int buf) {
        const unsigned gBase = (unsigned)(ch * CHUNK * DIM * 4);        // byte offset
        const unsigned lBase = (unsigned)(uintptr_t)&fSm[buf * CHUNK * FS];
        #pragma unroll
        for (int i = 0; i < 4; ++i) {
            const int j  = tid + i * 256;        // 0..1023 float4s
            const int r  = j >> 7;               // row (128 float4 per row)
            const int c4 = j & 127;
            const unsigned ldsOff = lBase + (unsigned)((r * FS + c4 * 4) * 4);
            const unsigned gOff   = gBase + (unsigned)(j * 16);
            asm volatile("global_load_async_to_lds_b128 %0, %1, %2"
                         :: "v"(ldsOff), "v"(gOff), "s"(fb)
                         : "memory");
        }
    };

    // ---- softmax for one chunk's 8 rows (threads 0..7, one row each) ----
    auto softmaxChunk = [&](int ch, int buf) {
        if (tid < CHUNK) {
            float* arow = &aSm[buf * CHUNK * AS + tid * AS];
            const float* srow = sb + (size_t)(ch * CHUNK + tid) * KG;
            float v[KG];
            float m = -3.402823e38f;
            #pragma unroll
            for (int k = 0; k < KG; ++k) { v[k] = srow[k]; m = fmaxf(m, v[k]); }
            float s = 0.0f;
            #pragma unroll
            for (int k = 0; k < KG; ++k) { v[k] = __expf(v[k] - m); s += v[k]; }
            const float inv = 1.0f / s;
            #pragma unroll
            for (int k = 0; k < KG; ++k)     arow[k] = v[k] * inv;
            #pragma unroll
            for (int k = KG; k < KPAD; ++k)  arow[k] = 0.0f;
        }
    };

    // ---- prologue: start chunk 0 ----
    issueChunk(0, 0);
    softmaxChunk(0, 0);

    for (int ch = 0; ch < NCHUNK; ++ch) {
        const int  cur      = ch & 1;
        const bool inflight = (ch + 1 < NCHUNK);

        // prefetch next chunk into the other buffer (its readers finished at
        // the trailing barrier of the previous iteration)
        if (inflight) {
            issueChunk(ch + 1, cur ^ 1);
            softmaxChunk(ch + 1, cur ^ 1);
        }

        // wait for *this* chunk's 4 async loads (in-order; the 4 just issued
        // may remain outstanding), then make it visible to all waves
        if (inflight) asm volatile("s_wait_asynccnt 0x4" ::: "memory");
        else          asm volatile("s_wait_asynccnt 0x0" ::: "memory");
        __syncthreads();

        const float* fbuf = &fSm[cur * CHUNK * FS];
        const float* abuf = &aSm[cur * CHUNK * AS];

        // ---- WMMA: full-EXEC, each wave owns 4 N-tiles of 16 columns ----
        #pragma unroll
        for (int kk = 0; kk < CHUNK; kk += 4) {
            const int rowA = kk + 2 * half;
            v2f a;
            a.x = abuf[rowA * AS + l15];            // A[m=l15][k = kk+2h]
            a.y = abuf[(rowA + 1) * AS + l15];      // A[m=l15][k = kk+2h+1]

            int db = wave * 16;
            v2f bb;
            bb.x = fbuf[rowA * FS + db + l15];
            bb.y = fbuf[(rowA + 1) * FS + db + l15];
            acc0 = __builtin_amdgcn_wmma_f32_16x16x4_f32(false, a, false, bb, (short)0, acc0, false, false);

            db = wave * 16 + 128;
            bb.x = fbuf[rowA * FS + db + l15];
            bb.y = fbuf[(rowA + 1) * FS + db + l15];
            acc1 = __builtin_amdgcn_wmma_f32_16x16x4_f32(false, a, false, bb, (short)0, acc1, false, false);

            db = wave * 16 + 256;
            bb.x = fbuf[rowA * FS + db + l15];
            bb.y = fbuf[(rowA + 1) * FS + db + l15];
            acc2 = __builtin_amdgcn_wmma_f32_16x16x4_f32(false, a, false, bb, (short)0, acc2, false, false);

            db = wave * 16 + 384;
            bb.x = fbuf[rowA * FS + db + l15];
            bb.y = fbuf[(rowA + 1) * FS + db + l15];
            acc3 = __builtin_amdgcn_wmma_f32_16x16x4_f32(false, a, false, bb, (short)0, acc3, false, false);
        }

        // ---- per-chunk a_sum accumulation (wave 0, lanes 0..15) ----
        if (tid < 16) {
            float s = 0.0f;
            #pragma unroll
            for (int r = 0; r < CHUNK; ++r) s += abuf[r * AS + tid];
            asumAcc += s;
        }
        __syncthreads();   // all reads of `cur` done before it is refilled
    }

    // ---- flush partial results: wf[m][n], C layout: m = v + 8*half, n = l15 ----
    float* wf = wsWf + (size_t)b * KPAD * DIM;
    #pragma unroll
    for (int v = 0; v < 8; ++v) {
        const int m = v + 8 * half;
        atomicAdd(&wf[(size_t)m * DIM + wave * 16 +   0 + l15], acc0[v]);
        atomicAdd(&wf[(size_t)m * DIM + wave * 16 + 128 + l15], acc1[v]);
        atomicAdd(&wf[(size_t)m * DIM + wave * 16 + 256 + l15], acc2[v]);
        atomicAdd(&wf[(size_t)m * DIM + wave * 16 + 384 + l15], acc3[v]);
    }
    if (tid < 16) atomicAdd(&wsAsum[b * KPAD + tid], asumAcc);
}

// =============================================================================
// Kernel 2: residual + L2 normalize.  One wave per (batch, center).
// =============================================================================
__global__ __launch_bounds__(256) void vlad_finalize_kernel(
    const float* __restrict__ cluster,   // [KG][D]
    const float* __restrict__ wsWf,      // [B][KPAD][D]
    const float* __restrict__ wsAsum,    // [B][KPAD]
    float* __restrict__ out)             // [B][KC*D]
{
    const int b    = blockIdx.x;
    const int tid  = threadIdx.x;
    const int k    = tid >> 5;           // center 0..7 (one wave each)
    const int lane = tid & 31;

    const float* wf = wsWf + ((size_t)b * KPAD + k) * DIM;
    const float  as = wsAsum[b * KPAD + k];
    const float* cl = cluster + (size_t)k * DIM;

    float r[16];
    float ss = 0.0f;
    #pragma unroll
    for (int i = 0; i < 16; ++i) {
        const int d = lane + i * 32;
        const float x = wf[d] - as * cl[d];
        r[i] = x;
        ss  += x * x;
    }
    // wave32 reduction
    #pragma unroll
    for (int off = 16; off >= 1; off >>= 1)
        ss += __shfl_xor(ss, off, 32);

    const float inv = 1.0f / sqrtf(fmaxf(ss, 1e-12f));
    float* ob = out + ((size_t)b * KC + k) * DIM;
    #pragma unroll
    for (int i = 0; i < 16; ++i)
        ob[lane + i * 32] = r[i] * inv;
}

// =============================================================================
extern "C" void kernel_launch(void* const* d_in, const int* in_sizes, int n_in,
                              void* d_out, int out_size, void* d_ws, size_t ws_size,
                              hipStream_t stream) {
    (void)in_sizes; (void)n_in; (void)out_size; (void)ws_size;
    const float* feat    = (const float*)d_in[0];
    const float* score   = (const float*)d_in[1];
    const float* cluster = (const float*)d_in[2];
    float* out = (float*)d_out;

    float* wsWf   = (float*)d_ws;                       // BATCH*KPAD*DIM floats
    float* wsAsum = wsWf + (size_t)BATCH * KPAD * DIM;  // BATCH*KPAD floats
    const size_t zeroBytes = ((size_t)BATCH * KPAD * DIM + (size_t)BATCH * KPAD) * sizeof(float);
    hipMemsetAsync(d_ws, 0, zeroBytes, stream);

    dim3 g1(BATCH, SPLIT);
    vlad_partial_kernel<<<g1, 256, 0, stream>>>(feat, score, wsWf, wsAsum);
    vlad_finalize_kernel<<<BATCH, 256, 0, stream>>>(cluster, wsWf, wsAsum, out);
}